// FNO2d_53532472377648
// MI455X (gfx1250) — compile-verified
//
#include <hip/hip_runtime.h>
#include <math.h>

typedef float v2f __attribute__((ext_vector_type(2)));
typedef float v8f __attribute__((ext_vector_type(8)));

#define HSZ (16ull*64ull*65536ull)   // one activation tensor [B=16][C=64][H*W=65536]

__device__ __forceinline__ v8f wmma4(v2f a, v2f b, v8f c) {
  // D = A(16x4 f32) * B(4x16 f32) + C(16x16 f32)
  return __builtin_amdgcn_wmma_f32_16x16x4_f32(false, a, false, b, (short)0, c, false, false);
}

__device__ __forceinline__ float gelu_exact(float x) {
  return 0.5f * x * (1.0f + erff(x * 0.7071067811865476f));
}

// ---------------- lift: h[b][c][pix] = x[b][pix] * fc0_w[c] + fc0_b[c] ----------------
__global__ void k_lift(const float* __restrict__ x, const float* __restrict__ w,
                       const float* __restrict__ bv, float* __restrict__ h) {
  unsigned t = blockIdx.x * 256u + threadIdx.x;   // < 2^26
  unsigned pix = t & 65535u;
  unsigned c = (t >> 16) & 63u;
  unsigned b = t >> 22;
  h[t] = x[(b << 16) + pix] * w[c] + bv[c];
}

// ---------------- twiddle tables (integer phase mod 256 keeps sincos exact) ----------------
__global__ void k_twiddle(float* __restrict__ Fw, float* __restrict__ Thre,
                          float* __restrict__ Thim, float* __restrict__ Gw) {
  int t = blockIdx.x * 256 + threadIdx.x;         // < 20480
  const float STEP = 6.283185307179586f / 256.0f;
  if (t < 8192) {                                 // Fw [w=256][col=32]; e^{-i 2pi kx w/256}
    int w = t >> 5, n = t & 31;
    float v = 0.0f;
    if (n < 12)      { int ph = (n * w) & 255;        v =  cosf(STEP * (float)ph); }
    else if (n < 24) { int ph = ((n - 12) * w) & 255; v = -sinf(STEP * (float)ph); }
    Fw[t] = v;
  } else if (t < 14336) {                         // Th [j=24][h=256]; e^{-i 2pi ky_j h/256}
    int u = t - 8192; int j = u >> 8, hh = u & 255;
    int ky = (j < 12) ? j : (232 + j);            // rows 0..11 and 244..255
    int ph = (ky * hh) & 255;
    float ang = STEP * (float)ph;
    Thre[u] = cosf(ang);
    Thim[u] = -sinf(ang);
  } else {                                        // Gw [k=24][w=256]; inverse + hermitian + 1/HW
    int u = t - 14336; int k = u >> 8, w = u & 255;
    int kx = (k < 12) ? k : (k - 12);
    float c = ((kx == 0) ? 1.0f : 2.0f) * (1.0f / 65536.0f);
    int ph = (kx * w) & 255;
    float ang = STEP * (float)ph;
    Gw[u] = (k < 12) ? c * cosf(ang) : -c * sinf(ang);
  }
}

// ---------------- stage A fwd: y[row=(b,c,h)][32] = h_row[256] @ Fw[256][32] (WMMA) ----------------
__global__ void k_dftW_fwd(const float* __restrict__ hin, const float* __restrict__ Fw,
                           float* __restrict__ y) {
  int lane = threadIdx.x;
  int r = lane & 15, g = lane >> 4;
  size_t m0 = (size_t)blockIdx.x * 16;
  int n0 = blockIdx.y * 16;
  const float* arow = hin + (m0 + r) * 256;
  v8f acc = {0,0,0,0,0,0,0,0};
  for (int k = 0; k < 256; k += 4) {
    int kk = k + 2 * g;
    v2f a; a.x = arow[kk];                 a.y = arow[kk + 1];
    v2f b; b.x = Fw[kk * 32 + n0 + r];     b.y = Fw[(kk + 1) * 32 + n0 + r];
    acc = wmma4(a, b, acc);
  }
  for (int v = 0; v < 8; ++v)
    y[(m0 + (size_t)(v + 8 * g)) * 32 + n0 + r] = acc[v];
}

// ---------------- stage B fwd: z[b,i,j,kx] = sum_h Th[j,h] * y[b,i,h,kx] (complex) ----------------
__global__ void k_dftH_fwd(const float* __restrict__ y, const float* __restrict__ Thre,
                           const float* __restrict__ Thim, float* __restrict__ z) {
  int t = blockIdx.x * 256 + threadIdx.x;  // < 294912
  int kx = t % 12; int j = (t / 12) % 24; int bi = t / 288;  // bi = b*64+i
  const float* yrow = y + (size_t)bi * 256 * 32;
  float re = 0.0f, im = 0.0f;
  for (int h = 0; h < 256; ++h) {
    float yr = yrow[h * 32 + kx];
    float yi = yrow[h * 32 + 12 + kx];
    float tr = Thre[j * 256 + h];
    float ti = Thim[j * 256 + h];
    re += yr * tr - yi * ti;
    im += yr * ti + yi * tr;
  }
  size_t o = ((size_t)bi * 288 + j * 12 + kx) * 2;
  z[o] = re; z[o + 1] = im;
}

// ---------------- mode mix: z2[b,o,mode] = sum_i z[b,i,mode] * W[i,o,mode] (complex WMMA) ----------------
__global__ void k_modemix(const float* __restrict__ z, const float* __restrict__ scw_l,
                          float* __restrict__ z2) {
  int lane = threadIdx.x;
  int r = lane & 15, g = lane >> 4;
  int mode = blockIdx.x;          // j*12+kx, 0..287
  int n0 = blockIdx.y * 16;       // o tile
  int half = (mode >= 144) ? 1 : 0;
  int m1 = (mode / 12) % 12;
  int m2 = mode % 12;
  // sc_w[l,half,i,o,m1,m2,comp] strides: comp=1, m2=2, m1=24, o=288, i=18432, half=1179648
  const float* wbase = scw_l + (size_t)half * 1179648ull + (size_t)m1 * 24 + (size_t)m2 * 2;
  v8f accR = {0,0,0,0,0,0,0,0};
  v8f accI = {0,0,0,0,0,0,0,0};
  for (int k = 0; k < 64; k += 4) {
    int i0 = k + 2 * g;
    size_t za = ((size_t)(r * 64 + i0) * 288 + mode) * 2;
    size_t zb = ((size_t)(r * 64 + i0 + 1) * 288 + mode) * 2;
    v2f zr, zi;
    zr.x = z[za];     zi.x = z[za + 1];
    zr.y = z[zb];     zi.y = z[zb + 1];
    size_t wa = (size_t)i0 * 18432 + (size_t)(n0 + r) * 288;
    size_t wc = wa + 18432;
    v2f wr, wi, win;
    wr.x = wbase[wa];     wi.x = wbase[wa + 1];
    wr.y = wbase[wc];     wi.y = wbase[wc + 1];
    win.x = -wi.x;        win.y = -wi.y;      // f32 WMMA has no A/B NEG modifier
    accR = wmma4(zr, wr, accR);
    accR = wmma4(zi, win, accR);
    accI = wmma4(zr, wi, accI);
    accI = wmma4(zi, wr, accI);
  }
  for (int v = 0; v < 8; ++v) {
    int bb = v + 8 * g;                       // batch row
    size_t oo = ((size_t)(bb * 64 + n0 + r) * 288 + mode) * 2;
    z2[oo] = accR[v]; z2[oo + 1] = accI[v];
  }
}

// ---------------- stage B inv: u[b,o,h,kx] = sum_j conj(Th[j,h]) * z2[b,o,j,kx] ----------------
__global__ void k_dftH_inv(const float* __restrict__ z2, const float* __restrict__ Thre,
                           const float* __restrict__ Thim, float* __restrict__ y) {
  int t = blockIdx.x * 256 + threadIdx.x;  // < 3145728
  int kx = t % 12; int h = (t / 12) & 255; int bo = t / 3072;  // bo = b*64+o
  float re = 0.0f, im = 0.0f;
  for (int j = 0; j < 24; ++j) {
    size_t zidx = ((size_t)bo * 288 + j * 12 + kx) * 2;
    float zr = z2[zidx], zi = z2[zidx + 1];
    float tr = Thre[j * 256 + h];
    float ti = -Thim[j * 256 + h];          // conjugate -> e^{+i...}
    re += zr * tr - zi * ti;
    im += zr * ti + zi * tr;
  }
  size_t row = (size_t)bo * 256 + h;
  y[row * 32 + kx]      = re;
  y[row * 32 + 12 + kx] = im;
}

// ---------------- fused layer tail (single write of h_next):
//   x2 = pw_w @ h (WMMA, K=64)           [rows = o, cols = 16 pixels of one h-row]
//   x1 = u[b,o,h_row, 0..23] @ Gw[:,w]   (VALU epilogue, K=24, inverse W-DFT)
//   h_next = gelu(x1 + x2 + pw_b)        (gelu skipped on last layer)
__global__ void k_pointwise_fused(const float* __restrict__ hin, const float* __restrict__ pw,
                                  const float* __restrict__ pb, const float* __restrict__ u,
                                  const float* __restrict__ Gw, float* __restrict__ hout,
                                  int apply_gelu) {
  int lane = threadIdx.x;
  int r = lane & 15, g = lane >> 4;
  size_t p0 = (size_t)blockIdx.x * 16;   // 16 consecutive pixels, same h-row
  int m0 = blockIdx.y * 16;              // o tile
  int b = blockIdx.z;
  const float* hb = hin + (size_t)b * 64 * 65536;
  v8f acc = {0,0,0,0,0,0,0,0};
  for (int k = 0; k < 64; k += 4) {
    int i0 = k + 2 * g;
    if (i0 + 4 < 64)
      __builtin_prefetch(hb + (size_t)(i0 + 4) * 65536 + p0 + r, 0, 0);
    v2f a;  a.x  = pw[(m0 + r) * 64 + i0];            a.y  = pw[(m0 + r) * 64 + i0 + 1];
    v2f bv; bv.x = hb[(size_t)i0 * 65536 + p0 + r];   bv.y = hb[(size_t)(i0 + 1) * 65536 + p0 + r];
    acc = wmma4(a, bv, acc);
  }
  // inverse W-DFT weights for this lane's column (w = w0 + r), K=24
  int hrow = (int)(p0 >> 8);             // h index of this pixel strip
  int w0 = (int)(p0 & 255);
  float gw[24];
  for (int k = 0; k < 24; ++k) gw[k] = Gw[k * 256 + w0 + r];
  for (int v = 0; v < 8; ++v) {
    int o = m0 + v + 8 * g;
    const float* urow = u + ((size_t)(b * 64 + o) * 256 + hrow) * 32;  // cols 0..11 re, 12..23 im
    float x1 = 0.0f;
    for (int k = 0; k < 24; ++k) x1 += urow[k] * gw[k];
    float val = acc[v] + pb[o] + x1;
    hout[((size_t)b * 64 + o) * 65536 + p0 + r] = apply_gelu ? gelu_exact(val) : val;
  }
}

// ---------------- head: out[p] = fc2( gelu( fc1(h[:,p]) ) ); fc1 via WMMA ----------------
__global__ void k_head(const float* __restrict__ hin, const float* __restrict__ w1,
                       const float* __restrict__ b1, const float* __restrict__ w2,
                       const float* __restrict__ b2, float* __restrict__ out) {
  __shared__ float red[16];
  int tid = threadIdx.x;
  int wave = tid >> 5, lane = tid & 31;
  int r = lane & 15, g = lane >> 4;
  size_t p0 = (size_t)blockIdx.x * 16;      // global pixel base (16 pixels, same b)
  int b = (int)(p0 >> 16);
  size_t pin = p0 & 65535;
  if (tid < 16) red[tid] = 0.0f;
  __syncthreads();
  int jb = wave * 16;                       // hidden-tile per wave: 8 waves x 16 = 128
  const float* hb = hin + (size_t)b * 64 * 65536 + pin;
  v8f acc = {0,0,0,0,0,0,0,0};
  for (int k = 0; k < 64; k += 4) {
    int c0 = k + 2 * g;
    v2f a;  a.x  = w1[c0 * 128 + jb + r];        a.y  = w1[(c0 + 1) * 128 + jb + r];
    v2f bv; bv.x = hb[(size_t)c0 * 65536 + r];   bv.y = hb[(size_t)(c0 + 1) * 65536 + r];
    acc = wmma4(a, bv, acc);                 // D rows = hidden j, cols = pixel
  }
  float s = 0.0f;
  for (int v = 0; v < 8; ++v) {
    int j = jb + v + 8 * g;
    float hv = gelu_exact(acc[v] + b1[j]);
    s += hv * w2[j];
  }
  atomicAdd(&red[r], s);                     // lane column r == pixel index
  __syncthreads();
  if (tid < 16) out[p0 + tid] = red[tid] + b2[0];
}

extern "C" void kernel_launch(void* const* d_in, const int* in_sizes, int n_in,
                              void* d_out, int out_size, void* d_ws, size_t ws_size,
                              hipStream_t stream) {
  (void)in_sizes; (void)n_in; (void)out_size; (void)ws_size;
  const float* x     = (const float*)d_in[0];
  const float* fc0_w = (const float*)d_in[1];
  const float* fc0_b = (const float*)d_in[2];
  const float* sc_w  = (const float*)d_in[3];
  const float* pw_w  = (const float*)d_in[4];
  const float* pw_b  = (const float*)d_in[5];
  const float* fc1_w = (const float*)d_in[6];
  const float* fc1_b = (const float*)d_in[7];
  const float* fc2_w = (const float*)d_in[8];
  const float* fc2_b = (const float*)d_in[9];
  float* out = (float*)d_out;

  float* ws   = (float*)d_ws;
  float* hA   = ws;                          // 67,108,864 floats
  float* hB   = hA + HSZ;                    // 67,108,864 floats
  float* ybuf = hB + HSZ;                    // 262144*32 = 8,388,608 (fwd y / inv u)
  float* zbuf = ybuf + 262144ull * 32;       // 589,824
  float* z2b  = zbuf + 589824ull;            // 589,824
  float* Fw   = z2b  + 589824ull;            // 8,192
  float* Thre = Fw   + 8192;                 // 6,144
  float* Thim = Thre + 6144;                 // 6,144
  float* Gw   = Thim + 6144;                 // 6,144

  k_twiddle<<<80, 256, 0, stream>>>(Fw, Thre, Thim, Gw);
  k_lift<<<262144, 256, 0, stream>>>(x, fc0_w, fc0_b, hA);

  float* cur = hA;
  float* nxt = hB;
  for (int l = 0; l < 4; ++l) {
    const float* scw_l = sc_w + (size_t)l * 2359296ull;
    const float* pw_l  = pw_w + (size_t)l * 4096;
    const float* pb_l  = pw_b + (size_t)l * 64;

    { dim3 g(16384, 2);     k_dftW_fwd<<<g, 32, 0, stream>>>(cur, Fw, ybuf); }
    k_dftH_fwd<<<1152, 256, 0, stream>>>(ybuf, Thre, Thim, zbuf);
    { dim3 g(288, 4);       k_modemix<<<g, 32, 0, stream>>>(zbuf, scw_l, z2b); }
    k_dftH_inv<<<12288, 256, 0, stream>>>(z2b, Thre, Thim, ybuf);
    { dim3 g(4096, 4, 16);  k_pointwise_fused<<<g, 32, 0, stream>>>(cur, pw_l, pb_l, ybuf, Gw,
                                                                    nxt, (l < 3) ? 1 : 0); }

    float* t = cur; cur = nxt; nxt = t;
  }

  k_head<<<65536, 256, 0, stream>>>(cur, fc1_w, fc1_b, fc2_w, fc2_b, out);
}